// MeshVAE_63642825392707
// MI455X (gfx1250) — compile-verified
//
#include <hip/hip_runtime.h>
#include <math.h>

typedef __attribute__((ext_vector_type(2))) float v2f;
typedef __attribute__((ext_vector_type(8))) float v8f;

static inline unsigned cdiv_u(long a, long b) { return (unsigned)((a + b - 1) / b); }

// ---------------- elementwise / utility kernels (float4, n % 4 == 0) ----------------

__global__ void k_zero4(float* __restrict__ p, long n4) {
  long i = (long)blockIdx.x * blockDim.x + threadIdx.x;
  if (i < n4) ((float4*)p)[i] = make_float4(0.f, 0.f, 0.f, 0.f);
}

__global__ void k_neg4(const float* __restrict__ a, float* __restrict__ o, long n4) {
  long i = (long)blockIdx.x * blockDim.x + threadIdx.x;
  if (i >= n4) return;
  float4 v = ((const float4*)a)[i];
  ((float4*)o)[i] = make_float4(-v.x, -v.y, -v.z, -v.w);
}

// dst[r][c] (R x Cp) = (r<R && c<C) ? src[r][c] : 0
__global__ void k_pad2d(const float* __restrict__ src, float* __restrict__ dst,
                        long R, int C, int Cp) {
  long i = (long)blockIdx.x * blockDim.x + threadIdx.x;
  if (i >= R * Cp) return;
  long r = i / Cp;
  int c = (int)(i % Cp);
  dst[i] = (c < C) ? src[r * C + c] : 0.0f;
}

// dst[k][f][o] (Kc x Finp x Foutp) from src[k][fin][fout], zero-padded
__global__ void k_padW(const float* __restrict__ src, float* __restrict__ dst,
                       int Kc, int Fin, int Finp, int Fout, int Foutp) {
  long i = (long)blockIdx.x * blockDim.x + threadIdx.x;
  long total = (long)Kc * Finp * Foutp;
  if (i >= total) return;
  int o = (int)(i % Foutp);
  long t = i / Foutp;
  int f = (int)(t % Finp);
  int k = (int)(t / Finp);
  dst[i] = (f < Fin && o < Fout) ? src[((long)k * Fin + f) * Fout + o] : 0.0f;
}

// dst[n][0..Cd) = src[n][0..Cd) with src row stride Cs
__global__ void k_compact(const float* __restrict__ src, float* __restrict__ dst,
                          long N, int Cs, int Cd) {
  long i = (long)blockIdx.x * blockDim.x + threadIdx.x;
  if (i >= N * Cd) return;
  long n = i / Cd;
  int c = (int)(i % Cd);
  dst[i] = src[n * Cs + c];
}

// ---------------- graph norm ----------------

__global__ void k_deg(const int* __restrict__ src, int nE, float* __restrict__ deg) {
  int e = blockIdx.x * blockDim.x + threadIdx.x;
  if (e < nE) atomicAdd(&deg[src[e]], 1.0f);
}

__global__ void k_norm(const int* __restrict__ src, const int* __restrict__ dst,
                       const float* __restrict__ deg, float* __restrict__ nrm, int nE) {
  int e = blockIdx.x * blockDim.x + threadIdx.x;
  if (e >= nE) return;
  float ds = deg[src[e]], dd = deg[dst[e]];
  float a = ds > 0.0f ? 1.0f / sqrtf(ds) : 0.0f;
  float c = dd > 0.0f ? 1.0f / sqrtf(dd) : 0.0f;
  nrm[e] = a * c;
}

// ---------------- sparse propagation (F % 4 == 0): out[dst,f] += scale*norm[e]*h[src,f] ----------------

__global__ void k_prop4(const float* __restrict__ h, const int* __restrict__ src,
                        const int* __restrict__ dst, const float* __restrict__ nrm,
                        long nE, int F, float scale, float* __restrict__ out) {
  long i = (long)blockIdx.x * blockDim.x + threadIdx.x;
  int q = F >> 2;
  if (i >= nE * q) return;
  long e = i / q;
  int f4 = (int)(i % q) << 2;
  float s = scale * nrm[e];
  float4 hv = *(const float4*)(h + (long)src[e] * F + f4);
  float* op = out + (long)dst[e] * F + f4;
  atomicAdd(op + 0, s * hv.x);
  atomicAdd(op + 1, s * hv.y);
  atomicAdd(op + 2, s * hv.z);
  atomicAdd(op + 3, s * hv.w);
}

// ---------------- barycentric pool (F % 4 == 0) ----------------

__global__ void k_pool4(const float* __restrict__ x, const int* __restrict__ idx,
                        const float* __restrict__ w, float* __restrict__ out,
                        long Nout, int F) {
  long i = (long)blockIdx.x * blockDim.x + threadIdx.x;
  int q = F >> 2;
  if (i >= Nout * q) return;
  long o = i / q;
  int f4 = (int)(i % q) << 2;
  float4 s = make_float4(0.f, 0.f, 0.f, 0.f);
#pragma unroll
  for (int k = 0; k < 3; ++k) {
    float wv = w[o * 3 + k];
    float4 xv = *(const float4*)(x + (long)idx[o * 3 + k] * F + f4);
    s.x = fmaf(wv, xv.x, s.x);
    s.y = fmaf(wv, xv.y, s.y);
    s.z = fmaf(wv, xv.z, s.z);
    s.w = fmaf(wv, xv.w, s.w);
  }
  *(float4*)(out + o * F + f4) = s;
}

// ---------------- fused Chebyshev WMMA GEMM ----------------
// C[M,Ncol] = [A0 | A1_slice0 .. A1_slice4] @ W[6*FIN, Ncol]  (+bias, +relu)
// Requirements: M%16==0, FIN%4==0, Ncol%16==0 -> NO per-lane guards anywhere.
// V_WMMA_F32_16X16X4_F32 lane layout (ISA 7.12.2, wave32):
//   A(16x4): lane L, vgpr j -> A[M=L&15][K=2*(L>>4)+j]   (contiguous pair -> b64 load)
//   B(4x16): lane L, vgpr j -> W[K=2*(L>>4)+j][N=L&15]
//   C/D:     lane L, vgpr v -> C[M=v+8*(L>>4)][N=L&15]

template <int FIN>
__global__ void __launch_bounds__(128)
k_gemm_cheb(const float* __restrict__ A0, const float* __restrict__ A1, long ss,
            const float* __restrict__ W, float* __restrict__ C,
            int M, int Ncol, const float* __restrict__ bias, int relu) {
  const int lane = threadIdx.x & 31;
  const int gw = blockIdx.x * (blockDim.x >> 5) + (threadIdx.x >> 5);
  const int tilesN = Ncol >> 4;
  if (gw >= (M >> 4) * tilesN) return;  // whole-wave uniform exit
  const int tm = gw / tilesN;
  const int tn = gw % tilesN;
  const int r = lane & 15;
  const int hh = lane >> 4;
  const int col = tn * 16 + r;

  v8f acc;
#pragma unroll
  for (int v = 0; v < 8; ++v) acc[v] = 0.0f;

  const long aoff = (long)(tm * 16 + r) * FIN + 2 * hh;
  const float* wp = W + col + (long)(2 * hh) * Ncol;  // advances 4 rows per k-step
  const long wstep = 4L * Ncol;

  for (int s = 0; s < 6; ++s) {
    const float* ap = (s == 0 ? A0 : A1 + (long)(s - 1) * ss) + aoff;
#pragma unroll
    for (int f = 0; f < FIN; f += 4) {
      v2f av = *(const v2f*)ap;  // A[row][k], A[row][k+1] : single b64 load
      v2f bv;
      bv[0] = wp[0];
      bv[1] = wp[Ncol];
      acc = __builtin_amdgcn_wmma_f32_16x16x4_f32(false, av, false, bv,
                                                  (short)0, acc, false, false);
      ap += 4;
      wp += wstep;
    }
  }

  float badd = bias ? bias[col] : 0.0f;  // uniform branch
#pragma unroll
  for (int v = 0; v < 8; ++v) {
    float vv = acc[v] + badd;
    if (relu) vv = fmaxf(vv, 0.0f);
    C[(long)(tm * 16 + v + 8 * hh) * Ncol + col] = vv;
  }
}

// ---------------- dense layers ----------------

// out[j] = relu(sum_i x[i]*W[i*64+j] + b[j]),  j = blockIdx.x in [0,64)
__global__ void k_dense64(const float* __restrict__ x, const float* __restrict__ W,
                          const float* __restrict__ b, float* __restrict__ out, int K) {
  __shared__ float red[256];
  int j = blockIdx.x;
  float s = 0.0f;
  for (int i = threadIdx.x; i < K; i += blockDim.x)
    s = fmaf(x[i], W[(long)i * 64 + j], s);
  red[threadIdx.x] = s;
  __syncthreads();
  for (int st = 128; st > 0; st >>= 1) {
    if ((int)threadIdx.x < st) red[threadIdx.x] += red[threadIdx.x + st];
    __syncthreads();
  }
  if (threadIdx.x == 0) out[j] = fmaxf(red[0] + b[j], 0.0f);
}

__global__ void k_z(const float* __restrict__ mu, const float* __restrict__ lv,
                    const float* __restrict__ eps, float* __restrict__ z) {
  int j = threadIdx.x;
  if (j < 64) z[j] = mu[j] + eps[j] * expf(0.5f * lv[j]);
}

// out[i] = relu(sum_j z[j]*W[j*NF+i] + b[i])
__global__ void k_lin(const float* __restrict__ z, const float* __restrict__ W,
                      const float* __restrict__ b, float* __restrict__ out, int NF) {
  int i = blockIdx.x * blockDim.x + threadIdx.x;
  if (i >= NF) return;
  float s = b[i];
#pragma unroll
  for (int j = 0; j < 64; ++j) s = fmaf(z[j], W[(long)j * NF + i], s);
  out[i] = fmaxf(s, 0.0f);
}

// ---------------- host orchestration ----------------

extern "C" void kernel_launch(void* const* d_in, const int* in_sizes, int n_in,
                              void* d_out, int out_size, void* d_ws, size_t ws_size,
                              hipStream_t stream) {
  (void)in_sizes; (void)n_in; (void)out_size; (void)ws_size;

  const int EN[4]  = {65536, 16384, 4096, 65536};
  const int NEc[4] = {6 * 65536, 6 * 16384, 6 * 4096, 6 * 65536};

  const float* x = (const float*)d_in[0];
  const int* edge[4] = {(const int*)d_in[1], (const int*)d_in[2],
                        (const int*)d_in[3], (const int*)d_in[4]};
  const int*   down_idx[3] = {(const int*)d_in[5],  (const int*)d_in[9],  (const int*)d_in[13]};
  const float* down_w[3]   = {(const float*)d_in[6],(const float*)d_in[10],(const float*)d_in[14]};
  const int*   up_idx[3]   = {(const int*)d_in[7],  (const int*)d_in[11], (const int*)d_in[15]};
  const float* up_w[3]     = {(const float*)d_in[8],(const float*)d_in[12],(const float*)d_in[16]};
  const float* encW[3] = {(const float*)d_in[17], (const float*)d_in[19], (const float*)d_in[21]};
  const float* encB[3] = {(const float*)d_in[18], (const float*)d_in[20], (const float*)d_in[22]};
  const float* decW[4] = {(const float*)d_in[23], (const float*)d_in[25],
                          (const float*)d_in[27], (const float*)d_in[29]};
  const float* decB[3] = {(const float*)d_in[24], (const float*)d_in[26], (const float*)d_in[28]};
  const float* muW = (const float*)d_in[30]; const float* muB = (const float*)d_in[31];
  const float* sgW = (const float*)d_in[32]; const float* sgB = (const float*)d_in[33];
  const float* lnW = (const float*)d_in[34]; const float* lnB = (const float*)d_in[35];
  const float* eps = (const float*)d_in[36];

  // ---- workspace layout (floats) ----
  float* ws = (float*)d_ws;
  long off = 0;
  auto alloc = [&](long n) { float* p = ws + off; off += n; return p; };
  float* normp[4];
  for (int g = 0; g < 4; ++g) normp[g] = alloc(NEc[g]);
  float* deg  = alloc(65536);
  float* Tall = alloc(5L * 65536 * 64);  // 5 Chebyshev slices, max N*Fin
  float* xp   = alloc((long)65536 * 4);  // padded input (Fin 3->4)
  float* o16  = alloc((long)65536 * 16); // padded final conv output (Fout 3->16)
  float* W0p  = alloc(6L * 4 * 32);      // enc_W0 padded
  float* W3p  = alloc(6L * 32 * 16);     // dec_W3 padded
  float* e0out = alloc((long)65536 * 32);
  float* h1    = alloc((long)16384 * 32);
  float* e1out = alloc((long)16384 * 64);
  float* h2    = alloc((long)4096 * 64);
  float* e2out = alloc((long)4096 * 128);
  float* flat  = alloc((long)1024 * 128);
  float* dlin  = alloc((long)1024 * 128);
  float* u2    = alloc((long)4096 * 128);
  float* d0out = alloc((long)4096 * 128);
  float* u1    = alloc((long)16384 * 128);
  float* d1out = alloc((long)16384 * 64);
  float* u0    = alloc((long)65536 * 64);
  float* d2out = alloc((long)65536 * 32);

  // ---- one-time per call: graph norms + padded weights ----
  for (int g = 0; g < 4; ++g) {
    k_zero4<<<cdiv_u(EN[g] / 4, 256), 256, 0, stream>>>(deg, EN[g] / 4);
    k_deg<<<cdiv_u(NEc[g], 256), 256, 0, stream>>>(edge[g], NEc[g], deg);
    k_norm<<<cdiv_u(NEc[g], 256), 256, 0, stream>>>(edge[g], edge[g] + NEc[g], deg,
                                                    normp[g], NEc[g]);
  }
  k_padW<<<cdiv_u(6L * 4 * 32, 256), 256, 0, stream>>>(encW[0], W0p, 6, 3, 4, 32, 32);
  k_padW<<<cdiv_u(6L * 32 * 16, 256), 256, 0, stream>>>(decW[3], W3p, 6, 32, 32, 3, 16);

  auto gemm_cheb = [&](const float* A0, long ss, const float* Wm, float* C,
                       int M, int Fin, int Ncol, const float* bias, int relu) {
    long tiles = (long)(M >> 4) * (Ncol >> 4);
    unsigned blocks = cdiv_u(tiles, 4);
    switch (Fin) {
      case 4:   k_gemm_cheb<4><<<blocks, 128, 0, stream>>>(A0, Tall, ss, Wm, C, M, Ncol, bias, relu); break;
      case 32:  k_gemm_cheb<32><<<blocks, 128, 0, stream>>>(A0, Tall, ss, Wm, C, M, Ncol, bias, relu); break;
      case 64:  k_gemm_cheb<64><<<blocks, 128, 0, stream>>>(A0, Tall, ss, Wm, C, M, Ncol, bias, relu); break;
      default:  k_gemm_cheb<128><<<blocks, 128, 0, stream>>>(A0, Tall, ss, Wm, C, M, Ncol, bias, relu); break;
    }
  };

  // Chebyshev conv: T0=in, T1=P*T0, Tk=2*P*T_{k-1}-T_{k-2}; slices T1..T5 in Tall.
  auto cheb = [&](const float* in, float* out, int g, int N, const float* Wc,
                  const float* bias, int Fin, int Fout, int relu) {
    const int* src = edge[g];
    const int* dst = edge[g] + NEc[g];
    const float* nm = normp[g];
    long ne = NEc[g];
    long ss = (long)N * Fin;
    unsigned pb = cdiv_u(ne * (Fin / 4), 256);
    unsigned eb = cdiv_u(ss / 4, 256);

    k_zero4<<<eb, 256, 0, stream>>>(Tall, ss / 4);                  // T1 = P*T0
    k_prop4<<<pb, 256, 0, stream>>>(in, src, dst, nm, ne, Fin, 1.0f, Tall);
    for (int k = 2; k < 6; ++k) {
      const float* pm2 = (k == 2) ? in : Tall + (long)(k - 3) * ss; // T_{k-2}
      const float* pm1 = Tall + (long)(k - 2) * ss;                 // T_{k-1}
      float* tk = Tall + (long)(k - 1) * ss;
      k_neg4<<<eb, 256, 0, stream>>>(pm2, tk, ss / 4);              // tk = -T_{k-2}
      k_prop4<<<pb, 256, 0, stream>>>(pm1, src, dst, nm, ne, Fin, 2.0f, tk);
    }
    gemm_cheb(in, ss, Wc, out, N, Fin, Fout, bias, relu);           // one fused GEMM
  };

  auto pool = [&](const float* xin, const int* idx, const float* w, float* outp,
                  long Nout, int F) {
    k_pool4<<<cdiv_u(Nout * (F / 4), 256), 256, 0, stream>>>(xin, idx, w, outp, Nout, F);
  };

  float* out = (float*)d_out;
  const long RECON = (long)8 * 65536 * 3;

  for (int b = 0; b < 8; ++b) {
    const float* hin = x + (long)b * 65536 * 3;

    // encoder
    k_pad2d<<<cdiv_u((long)65536 * 4, 256), 256, 0, stream>>>(hin, xp, 65536, 3, 4);
    cheb(xp, e0out, 0, 65536, W0p, encB[0], 4, 32, 1);
    pool(e0out, down_idx[0], down_w[0], h1, 16384, 32);
    cheb(h1, e1out, 1, 16384, encW[1], encB[1], 32, 64, 1);
    pool(e1out, down_idx[1], down_w[1], h2, 4096, 64);
    cheb(h2, e2out, 2, 4096, encW[2], encB[2], 64, 128, 1);
    pool(e2out, down_idx[2], down_w[2], flat, 1024, 128);

    // latent
    float* mu = out + RECON + (long)b * 64;
    float* lv = out + RECON + 512 + (long)b * 64;
    float* zz = out + RECON + 1024 + (long)b * 64;
    k_dense64<<<64, 256, 0, stream>>>(flat, muW, muB, mu, 131072);
    k_dense64<<<64, 256, 0, stream>>>(flat, sgW, sgB, lv, 131072);
    k_z<<<1, 64, 0, stream>>>(mu, lv, eps + (long)b * 64, zz);
    k_lin<<<cdiv_u(131072, 256), 256, 0, stream>>>(zz, lnW, lnB, dlin, 131072);

    // decoder
    pool(dlin, up_idx[2], up_w[2], u2, 4096, 128);
    cheb(u2, d0out, 2, 4096, decW[0], decB[0], 128, 128, 1);
    pool(d0out, up_idx[1], up_w[1], u1, 16384, 128);
    cheb(u1, d1out, 1, 16384, decW[1], decB[1], 128, 64, 1);
    pool(d1out, up_idx[0], up_w[0], u0, 65536, 64);
    cheb(u0, d2out, 0, 65536, decW[2], decB[2], 64, 32, 1);
    cheb(d2out, o16, 3, 65536, W3p, nullptr, 32, 16, 0);
    k_compact<<<cdiv_u((long)65536 * 3, 256), 256, 0, stream>>>(
        o16, out + (long)b * 65536 * 3, 65536, 16, 3);
  }
}